// OnsetEdgePoolingVersion2_4750233830165
// MI455X (gfx1250) — compile-verified
//
#include <hip/hip_runtime.h>

typedef float v2f __attribute__((ext_vector_type(2)));
typedef float v8f __attribute__((ext_vector_type(8)));

#define DIM     128
#define NNODES  50000
#define NEDGES  800000
#define NKEEP   25000

// ---------------------------------------------------------------------------
// Kernel 1: tx = x @ W^T + b  via V_WMMA_F32_16X16X4_F32, one wave per 16-row
// strip, all 8 column tiles per wave (A fragment reused 8x per K-step).
// Also writes sums = tx (self-loop contribution folded into the init).
// ---------------------------------------------------------------------------
__global__ __launch_bounds__(32) void gemm_wmma_f32(
    const float* __restrict__ x, const float* __restrict__ W,
    const float* __restrict__ bias, float* __restrict__ tx,
    float* __restrict__ sums) {
  const int rowBase = blockIdx.x * 16;       // 3125 blocks, exact
  const int lane = threadIdx.x;              // 0..31, EXEC all ones
  const int half = lane >> 4;                // K-half select
  const int lcol = lane & 15;                // M for A, N for B

  v8f acc[8];
#pragma unroll
  for (int n = 0; n < 8; ++n) acc[n] = (v8f)(0.0f);

  // A: 16x4 fp32 tile of x. lane<16 -> K={k0,k0+1}, lane>=16 -> K={k0+2,k0+3}
  const float* xrow = x + (size_t)(rowBase + lcol) * DIM;

  for (int k0 = 0; k0 < DIM; k0 += 4) {
    const int k = k0 + half * 2;
    v2f a;
    a.x = xrow[k];
    a.y = xrow[k + 1];
#pragma unroll
    for (int n = 0; n < 8; ++n) {
      // B[k][ncol] = W[ncol][k]  (row-major W -> contiguous float2 load)
      const float* wrow = W + (size_t)(n * 16 + lcol) * DIM + k;
      v2f bf;
      bf.x = wrow[0];
      bf.y = wrow[1];
      acc[n] = __builtin_amdgcn_wmma_f32_16x16x4_f32(
          /*neg_a=*/false, a, /*neg_b=*/false, bf,
          /*c_mod=*/(short)0, acc[n], /*reuse_a=*/false, /*reuse_b=*/false);
    }
  }

  // C/D layout: VGPR r, lanes 0-15 -> M=r, lanes 16-31 -> M=r+8; N = lcol
#pragma unroll
  for (int n = 0; n < 8; ++n) {
    const int col = n * 16 + lcol;
    const float bv = bias[col];
#pragma unroll
    for (int r = 0; r < 8; ++r) {
      const int row = rowBase + r + half * 8;
      const float v = acc[n][r] + bv;
      tx[(size_t)row * DIM + col] = v;
      sums[(size_t)row * DIM + col] = v;  // self-loop: sums starts at tx
    }
  }
}

// ---------------------------------------------------------------------------
// Kernel 2: cnt = 1.0 (self-loop count)
// ---------------------------------------------------------------------------
__global__ __launch_bounds__(256) void cnt_init(float* __restrict__ cnt) {
  const int i = blockIdx.x * blockDim.x + threadIdx.x;
  if (i < NNODES) cnt[i] = 1.0f;
}

// ---------------------------------------------------------------------------
// Kernel 3: edge scatter-add. One wave32 per edge: each lane moves float4
// (128 floats / 32 lanes) with global_atomic_add_f32 into L2-resident sums.
// ---------------------------------------------------------------------------
__global__ __launch_bounds__(256) void edge_scatter(
    const int* __restrict__ ei, const float* __restrict__ tx,
    float* __restrict__ sums, float* __restrict__ cnt) {
  const int gid = blockIdx.x * blockDim.x + threadIdx.x;
  const int e = gid >> 5;
  const int lane = gid & 31;
  if (e >= NEDGES) return;
  const int src = ei[e];
  const int dst = ei[NEDGES + e];

  const float4 v = ((const float4*)(tx + (size_t)src * DIM))[lane];
  float* srow = sums + (size_t)dst * DIM + lane * 4;
  atomicAdd(srow + 0, v.x);
  atomicAdd(srow + 1, v.y);
  atomicAdd(srow + 2, v.z);
  atomicAdd(srow + 3, v.w);
  if (lane == 0) atomicAdd(cnt + dst, 1.0f);
}

// ---------------------------------------------------------------------------
// Kernel 4: h = sums/cnt, gather rows by idx; append idx (as float) after the
// [NKEEP, DIM] block to match the flat concatenated tuple output.
// ---------------------------------------------------------------------------
__global__ __launch_bounds__(DIM) void gather_out(
    const float* __restrict__ sums, const float* __restrict__ cnt,
    const int* __restrict__ idx, float* __restrict__ out) {
  const int i = blockIdx.x;   // 0..NKEEP-1
  const int d = threadIdx.x;  // 0..127
  const int r = idx[i];
  const float c = cnt[r];
  out[(size_t)i * DIM + d] = sums[(size_t)r * DIM + d] / c;
  if (d == 0) out[(size_t)NKEEP * DIM + i] = (float)r;
}

extern "C" void kernel_launch(void* const* d_in, const int* in_sizes, int n_in,
                              void* d_out, int out_size, void* d_ws,
                              size_t ws_size, hipStream_t stream) {
  const float* x   = (const float*)d_in[0];  // [50000,128]
  const int*   ei  = (const int*)d_in[1];    // [2,800000]
  const int*   idx = (const int*)d_in[2];    // [25000]
  const float* W   = (const float*)d_in[3];  // [128,128]
  const float* b   = (const float*)d_in[4];  // [128]

  float* tx   = (float*)d_ws;                         // 50000*128
  float* sums = tx + (size_t)NNODES * DIM;            // 50000*128
  float* cnt  = sums + (size_t)NNODES * DIM;          // 50000

  gemm_wmma_f32<<<NNODES / 16, 32, 0, stream>>>(x, W, b, tx, sums);
  cnt_init<<<(NNODES + 255) / 256, 256, 0, stream>>>(cnt);
  edge_scatter<<<(NEDGES * 32 + 255) / 256, 256, 0, stream>>>(ei, tx, sums, cnt);
  gather_out<<<NKEEP, DIM, 0, stream>>>(sums, cnt, idx, (float*)d_out);
}